// StructNConv2D_gy_with_ds_22660247454227
// MI455X (gfx1250) — compile-verified
//
#include <hip/hip_runtime.h>
#include <hip/hip_bf16.h>

#define EPS 1e-20f

// Problem constants from the reference
#define B_  2
#define C_  32
#define O_  32
#define H_  256
#define W_  512
#define HO_ 254
#define WO_ 510
#define PIX_ (B_ * HO_ * WO_)          // 259080 pixels
#define OUTHALF_ (B_ * O_ * HO_ * WO_) // 8290560 elements per output tensor

typedef float v2f __attribute__((ext_vector_type(2)));
typedef float v8f __attribute__((ext_vector_type(8)));

__device__ __forceinline__ float softplus_f(float x) {
    // matches jax.nn.softplus = logaddexp(x, 0)
    return fmaxf(x, 0.f) + log1pf(expf(-fabsf(x)));
}

// -------- Kernel 1: vertical propagation (roll-based) --------
__global__ void prop_gy_kernel(const float* __restrict__ d,
                               const float* __restrict__ cd,
                               const float* __restrict__ s,
                               const float* __restrict__ gy,
                               const float* __restrict__ cgy,
                               const float* __restrict__ w_prop,
                               float* __restrict__ gy_f,
                               float* __restrict__ cgy_f,
                               int total) {
    int idx = blockIdx.x * blockDim.x + threadIdx.x;
    if (idx >= total) return;

    int ww = idx % W_;
    int h  = (idx / W_) % H_;
    int bc = idx / (W_ * H_);
    int base = bc * H_ * W_;

    int hu = (h == 0) ? (H_ - 1) : (h - 1);      // jnp.roll wraps
    int hd = (h == H_ - 1) ? 0 : (h + 1);

    float wprop = softplus_f(w_prop[0]);

    float d_up  = d[base + hu * W_ + ww];
    float d_dn  = d[base + hd * W_ + ww];
    float cd_up = (h == 0)      ? 0.f : cd[base + hu * W_ + ww];
    float cd_dn = (h == H_ - 1) ? 0.f : cd[base + hd * W_ + ww];
    float s_c   = s[idx];
    float s_up  = s[base + hu * W_ + ww];
    float s_dn  = s[base + hd * W_ + ww];
    float gyv   = gy[idx];
    float cgyv  = cgy[idx];

    float cgy_ds = s_c * s_up * s_dn * cd_up * cd_up;   // faithful: cd_up twice
    float height = (cd_up * d_up + cd_dn * d_dn) / (cd_up + cd_dn + EPS);
    float gy_ds  = (d_dn - d_up) * 0.5f / (height + EPS);

    float denf = wprop * cgyv + cgy_ds;
    gy_f[idx]  = (wprop * cgyv * gyv + cgy_ds * gy_ds) / (denf + EPS);
    cgy_f[idx] = denf / (wprop + 1.f);
}

// -------- Kernel 2: fused 3x3 spatial conv + 1x1 channel mix (WMMA) --------
// 256 threads = 8 waves per block; each wave owns a tile of 32 pixels.
__global__ void spatial_channel_kernel(const float* __restrict__ gy_f,
                                       const float* __restrict__ cgy_f,
                                       const float* __restrict__ sprod,
                                       const float* __restrict__ sw_raw,   // (C,1,9,1,1)
                                       const float* __restrict__ cw_raw,   // (O,C,1,1)
                                       const float* __restrict__ bias,     // (1,O,1,1)
                                       float* __restrict__ out) {
    __shared__ float s_cw[O_ * C_];       // softplus(channel_weight), [o][c]
    __shared__ float s_sw[C_ * 9];        // softplus(spatial_weight), [c][k]
    __shared__ float s_bias[O_];
    __shared__ float s_sums[2];           // {sum(sw), sum(cw)}
    __shared__ float s_X[8][2][C_][33];   // per-wave: [0]=cgy_sp*gy_sp, [1]=cgy_sp

    const int tid  = threadIdx.x;
    const int wave = tid >> 5;
    const int lane = tid & 31;

    // ---- Phase 1: parameter preprocessing ----
    for (int i = tid; i < O_ * C_; i += 256) s_cw[i] = softplus_f(cw_raw[i]);
    for (int i = tid; i < C_ * 9; i += 256)  s_sw[i] = softplus_f(sw_raw[i]);
    if (tid < O_) s_bias[tid] = bias[tid];
    __syncthreads();
    if (tid == 0) { float t = 0.f; for (int i = 0; i < C_ * 9; ++i) t += s_sw[i]; s_sums[0] = t; }
    if (tid == 1) { float t = 0.f; for (int i = 0; i < O_ * C_; ++i) t += s_cw[i]; s_sums[1] = t; }
    __syncthreads();

    const float sum_sw = s_sums[0];
    const float sum_cw = s_sums[1];

    // ---- Phase 2: per-(channel,pixel) spatial reduction into LDS ----
    const int tile = blockIdx.x * 8 + wave;
    const int p = tile * 32 + lane;
    const bool valid = (p < PIX_);
    int b = 0, ho = 0, wo = 0;
    if (valid) {
        b = p / (HO_ * WO_);
        int r = p - b * (HO_ * WO_);
        ho = r / WO_;
        wo = r - ho * WO_;
    }

    for (int c = 0; c < C_; ++c) {
        float nom = 0.f, den = 0.f;
        if (valid) {
            const int gbase = ((b * C_ + c) * H_ + ho) * W_ + wo;
            const int pbase = ((b * C_ + c) * 9) * (HO_ * WO_) + ho * WO_ + wo;
            #pragma unroll
            for (int i = 0; i < 3; ++i) {
                #pragma unroll
                for (int j = 0; j < 3; ++j) {
                    const int k = i * 3 + j;
                    float g  = gy_f[gbase + i * W_ + j];
                    float cg = cgy_f[gbase + i * W_ + j];
                    float sp = sprod[pbase + k * (HO_ * WO_)];
                    float cp = cg * sp;
                    float wk = s_sw[c * 9 + k];
                    nom += cp * g * wk;
                    den += cp * wk;
                }
            }
        }
        float gy_sp  = nom / (den + EPS);
        float cgy_sp = den / sum_sw;
        s_X[wave][0][c][lane] = valid ? (cgy_sp * gy_sp) : 0.f;
        s_X[wave][1][c][lane] = valid ? cgy_sp : 0.f;
    }
    __syncthreads();

    // ---- Phase 3: channel mix via V_WMMA_F32_16X16X4_F32 ----
    // D[o,p] = sum_c cw[o,c] * X[c,p]; M=o, N=p, K=c (8 steps of 4)
    const int mlane   = lane & 15;
    const int kparity = (lane >> 4) * 2;  // A: VGPR0 holds K=0/2; B mirrors
    const int madd    = (lane >> 4) * 8;  // C/D: lanes>=16 hold M=r+8

    for (int mt = 0; mt < 2; ++mt) {
        const int o_base = mt * 16;
        const int mrow = o_base + mlane;
        for (int nt = 0; nt < 2; ++nt) {
            const int p_off = nt * 16;
            const int col = p_off + mlane;
            v8f accN = {};
            v8f accD = {};
            #pragma unroll
            for (int kk = 0; kk < 8; ++kk) {
                const int kc = kk * 4 + kparity;
                v2f a, bn, bd;
                a.x = s_cw[mrow * C_ + kc];
                a.y = s_cw[mrow * C_ + kc + 1];
                bn.x = s_X[wave][0][kc][col];
                bn.y = s_X[wave][0][kc + 1][col];
                bd.x = s_X[wave][1][kc][col];
                bd.y = s_X[wave][1][kc + 1][col];
                accN = __builtin_amdgcn_wmma_f32_16x16x4_f32(
                    false, a, false, bn, (short)0, accN, false, false);
                accD = __builtin_amdgcn_wmma_f32_16x16x4_f32(
                    false, a, false, bd, (short)0, accD, false, false);
            }
            // store results: acc[r] holds D[M = r + madd][N = mlane]
            const int pp = tile * 32 + p_off + mlane;
            if (pp < PIX_) {
                int b2 = pp / (HO_ * WO_);
                int r2 = pp - b2 * (HO_ * WO_);
                int ho2 = r2 / WO_;
                int wo2 = r2 - ho2 * WO_;
                #pragma unroll
                for (int r = 0; r < 8; ++r) {
                    int o = o_base + r + madd;
                    int oidx = ((b2 * O_ + o) * HO_ + ho2) * WO_ + wo2;
                    float nom2 = accN[r];
                    float den2 = accD[r];
                    out[oidx]            = nom2 / (den2 + EPS) + s_bias[o];
                    out[OUTHALF_ + oidx] = den2 / sum_cw;
                }
            }
        }
    }
}

extern "C" void kernel_launch(void* const* d_in, const int* in_sizes, int n_in,
                              void* d_out, int out_size, void* d_ws, size_t ws_size,
                              hipStream_t stream) {
    const float* d      = (const float*)d_in[0];
    const float* cd     = (const float*)d_in[1];
    const float* s      = (const float*)d_in[2];
    // d_in[3..5] = cs, gx, cgx : unused by the reference's gy path
    const float* gy     = (const float*)d_in[6];
    const float* cgy    = (const float*)d_in[7];
    const float* sprod  = (const float*)d_in[8];
    const float* w_prop = (const float*)d_in[9];
    const float* sw_raw = (const float*)d_in[10];
    const float* cw_raw = (const float*)d_in[11];
    const float* bias   = (const float*)d_in[12];
    float* out = (float*)d_out;

    float* ws    = (float*)d_ws;
    const int N  = B_ * C_ * H_ * W_;    // 8388608
    float* gy_f  = ws;
    float* cgy_f = ws + N;

    prop_gy_kernel<<<(N + 255) / 256, 256, 0, stream>>>(
        d, cd, s, gy, cgy, w_prop, gy_f, cgy_f, N);

    const int tiles  = (PIX_ + 31) / 32;     // 8097
    const int blocks = (tiles + 7) / 8;      // 1013
    spatial_channel_kernel<<<blocks, 256, 0, stream>>>(
        gy_f, cgy_f, sprod, sw_raw, cw_raw, bias, out);
}